// DiffPool_65420941852759
// MI455X (gfx1250) — compile-verified
//
#include <hip/hip_runtime.h>
#include <hip/hip_bf16.h>

typedef __attribute__((ext_vector_type(2))) float v2f;
typedef __attribute__((ext_vector_type(8))) float v8f;

#define WG 256
#define NTPW 4   // N-tiles (16 cols each) per wave

// ---------------------------------------------------------------- utilities
__global__ void k_fill(float* __restrict__ p, long long n, float val) {
  long long i = (long long)blockIdx.x * blockDim.x + threadIdx.x;
  long long stride = (long long)gridDim.x * blockDim.x;
  for (; i < n; i += stride) p[i] = val;
}

__global__ void k_deg_edges(const int* __restrict__ row, float* __restrict__ deg, int E) {
  int stride = gridDim.x * blockDim.x;
  for (int e = blockIdx.x * blockDim.x + threadIdx.x; e < E; e += stride)
    atomicAdd(&deg[row[e]], 1.0f);
}

__global__ void k_dinv(const float* __restrict__ deg, float* __restrict__ dinv, int N) {
  int stride = gridDim.x * blockDim.x;
  for (int i = blockIdx.x * blockDim.x + threadIdx.x; i < N; i += stride) {
    float d = deg[i];
    dinv[i] = d > 0.0f ? rsqrtf(d) : 0.0f;
  }
}

// ------------------------------------------- shared WMMA pipeline helpers
// A fragment: a[v] = A[abase + (kk+v)*sa]   (sa = 1 for row-major X, M for S^T)
// B fragment: b[t][v] = Bm[bbase[t] + (kk+v)*sb]
__device__ __forceinline__ void load_stage(const float* __restrict__ A,
                                           const float* __restrict__ Bm,
                                           unsigned abase, unsigned sa,
                                           const unsigned* bbase, unsigned sb,
                                           int kk, v2f& a, v2f* b) {
  const unsigned k0 = (unsigned)kk;
  a[0] = A[abase + k0 * sa];
  a[1] = A[abase + (k0 + 1u) * sa];
#pragma unroll
  for (int t = 0; t < NTPW; ++t) {
    b[t][0] = Bm[bbase[t] + k0 * sb];
    b[t][1] = Bm[bbase[t] + (k0 + 1u) * sb];
  }
}

__device__ __forceinline__ void mma_stage(const v2f& a, const v2f* b, v8f* acc) {
#pragma unroll
  for (int t = 0; t < NTPW; ++t)
    acc[t] = __builtin_amdgcn_wmma_f32_16x16x4_f32(false, a, false, b[t], (short)0,
                                                   acc[t], false, false);
}

// ------------------------------------------------- H = X @ W + b  (fp32 WMMA)
// X:[N,D], W:[D,OD], H:[N,OD] row-major. 16-row x (NTPW*16)-col strip per wave;
// branch-free clamped loads, 32-bit offsets, software-pipelined K loop.
__global__ void k_linear_wmma(const float* __restrict__ X, const float* __restrict__ W,
                              const float* __restrict__ bias, float* __restrict__ H,
                              int N, int D, int OD, int ngroups) {
  const int wid = threadIdx.x >> 5;
  const int lane = threadIdx.x & 31;
  const int half = lane >> 4;
  const int l16 = lane & 15;
  const int mtiles = (N + 15) >> 4;
  int wunit = blockIdx.x * (blockDim.x >> 5) + wid;
  if (wunit >= mtiles * ngroups) return;               // wave-uniform exit
  const int mt = wunit / ngroups;
  const int ntg = (wunit % ngroups) * NTPW;

  const int m = mt * 16 + l16;
  const unsigned abase = (unsigned)min(m, N - 1) * (unsigned)D + (unsigned)(half * 2);
  int ncol[NTPW];
  unsigned bbase[NTPW];
#pragma unroll
  for (int t = 0; t < NTPW; ++t) {
    ncol[t] = (ntg + t) * 16 + l16;
    bbase[t] = (unsigned)(half * 2) * (unsigned)OD + (unsigned)min(ncol[t], OD - 1);
  }

  v8f acc[NTPW] = {};
  {                                                    // pipelined: D % 4 == 0, D >= 4
    v2f a, b[NTPW];
    load_stage(X, W, abase, 1u, bbase, (unsigned)OD, 0, a, b);
#pragma unroll 2
    for (int kk = 4; kk < D; kk += 4) {
      v2f an, bn[NTPW];
      load_stage(X, W, abase, 1u, bbase, (unsigned)OD, kk, an, bn);  // next in flight
      mma_stage(a, b, acc);                                          // compute current
      a = an;
#pragma unroll
      for (int t = 0; t < NTPW; ++t) b[t] = bn[t];
    }
    mma_stage(a, b, acc);                              // epilogue
  }

#pragma unroll
  for (int t = 0; t < NTPW; ++t) {
    if (ncol[t] < OD) {
      const float bv = bias[ncol[t]];
#pragma unroll
      for (int v = 0; v < 8; ++v) {
        const int m2 = mt * 16 + v + half * 8;         // D-layout: M = v + 8*half
        if (m2 < N) H[(unsigned)m2 * (unsigned)OD + (unsigned)ncol[t]] = acc[t][v] + bv;
      }
    }
  }
}

// ----------------------------------- self-loop term: OUT = dinv^2 * H, wave/row
__global__ void k_gcn_init(const float* __restrict__ H, const float* __restrict__ dinv,
                           float* __restrict__ OUT, int N, int dim) {
  const int wid = threadIdx.x >> 5;
  const int lane = threadIdx.x & 31;
  const int waves = (gridDim.x * blockDim.x) >> 5;
  for (int node = blockIdx.x * (blockDim.x >> 5) + wid; node < N; node += waves) {
    const float d2 = dinv[node] * dinv[node];
    const unsigned base = (unsigned)node * (unsigned)dim;
    for (int j = lane; j < dim; j += 32) OUT[base + j] = d2 * H[base + j];
  }
}

// ------------------- edge scatter: OUT[sidx[e]] += (norm?) * SRC[gidx[e]] rowwise
// idx = e*GROUPS + g (g fastest) -> coalesced float4 along the feature dim.
// GROUPS as template constant so idx/GROUPS strength-reduces to mul-shift.
template <int GROUPS>
__global__ void k_edge_scatter_t(const int* __restrict__ gidx, const int* __restrict__ sidx,
                                 const float* __restrict__ dinv, const float* __restrict__ SRC,
                                 float* __restrict__ OUT, int E, int use_norm) {
  const int dim = GROUPS * 4;
  const unsigned total = (unsigned)E * (unsigned)GROUPS;
  const unsigned stride = gridDim.x * blockDim.x;
  for (unsigned idx = blockIdx.x * blockDim.x + threadIdx.x; idx < total; idx += stride) {
    const unsigned e = idx / (unsigned)GROUPS;
    const unsigned g = idx - e * (unsigned)GROUPS;
    const int gi = gidx[e];
    const int si = sidx[e];
    const float nrm = use_norm ? dinv[gi] * dinv[si] : 1.0f;
    const float4 h = *(const float4*)(SRC + (unsigned)gi * (unsigned)dim + 4u * g);
    float* o = OUT + (unsigned)si * (unsigned)dim + 4u * g;
    atomicAdd(o + 0, nrm * h.x);
    atomicAdd(o + 1, nrm * h.y);
    atomicAdd(o + 2, nrm * h.z);
    atomicAdd(o + 3, nrm * h.w);
  }
}

__global__ void k_edge_scatter_g(const int* __restrict__ gidx, const int* __restrict__ sidx,
                                 const float* __restrict__ dinv, const float* __restrict__ SRC,
                                 float* __restrict__ OUT, int E, int dim, int use_norm) {
  const unsigned groups = (unsigned)(dim >> 2);
  const unsigned total = (unsigned)E * groups;
  const unsigned stride = gridDim.x * blockDim.x;
  for (unsigned idx = blockIdx.x * blockDim.x + threadIdx.x; idx < total; idx += stride) {
    const unsigned e = idx / groups;
    const unsigned g = idx - e * groups;
    const int gi = gidx[e];
    const int si = sidx[e];
    const float nrm = use_norm ? dinv[gi] * dinv[si] : 1.0f;
    const float4 h = *(const float4*)(SRC + (unsigned)gi * (unsigned)dim + 4u * g);
    float* o = OUT + (unsigned)si * (unsigned)dim + 4u * g;
    atomicAdd(o + 0, nrm * h.x);
    atomicAdd(o + 1, nrm * h.y);
    atomicAdd(o + 2, nrm * h.z);
    atomicAdd(o + 3, nrm * h.w);
  }
}

// ------------------------------------------------- row softmax, one wave per row
__global__ void k_softmax(float* __restrict__ S, int N, int K) {
  const int wid = threadIdx.x >> 5;
  const int lane = threadIdx.x & 31;
  const int row = blockIdx.x * (blockDim.x >> 5) + wid;
  if (row >= N) return;                                // wave-uniform
  float* p = S + (unsigned)row * (unsigned)K;
  float mx = -3.402823466e+38f;
  for (int j = lane; j < K; j += 32) mx = fmaxf(mx, p[j]);
#pragma unroll
  for (int off = 16; off > 0; off >>= 1) mx = fmaxf(mx, __shfl_xor(mx, off, 32));
  float s = 0.0f;
  for (int j = lane; j < K; j += 32) {
    float e = expf(p[j] - mx);
    p[j] = e;
    s += e;
  }
#pragma unroll
  for (int off = 16; off > 0; off >>= 1) s += __shfl_xor(s, off, 32);
  const float inv = 1.0f / s;
  for (int j = lane; j < K; j += 32) p[j] *= inv;
}

// --------------------------- OUT[M,NB] += S^T @ B, split-K over node chunks (WMMA)
// S:[N,M] (A^T source), B:[N,NB]. blockIdx.x = K chunk, blockIdx.y covers units.
// Branch-free clamped loads, software-pipelined main loop, guarded K tail.
__global__ void k_spt_gemm_wmma(const float* __restrict__ S, const float* __restrict__ B,
                                float* __restrict__ OUT, int N, int M, int NB,
                                int KC, int ngroups) {
  const int wid = threadIdx.x >> 5;
  const int lane = threadIdx.x & 31;
  const int half = lane >> 4;
  const int l16 = lane & 15;
  const int mtiles = (M + 15) >> 4;
  int wunit = blockIdx.y * (blockDim.x >> 5) + wid;
  if (wunit >= mtiles * ngroups) return;               // wave-uniform exit
  const int mt = wunit / ngroups;
  const int ntg = (wunit % ngroups) * NTPW;

  const int mcol = mt * 16 + l16;                      // A row == S column
  const unsigned abase = (unsigned)(half * 2) * (unsigned)M + (unsigned)min(mcol, M - 1);
  int ncol[NTPW];
  unsigned bbase[NTPW];
#pragma unroll
  for (int t = 0; t < NTPW; ++t) {
    ncol[t] = (ntg + t) * 16 + l16;
    bbase[t] = (unsigned)(half * 2) * (unsigned)NB + (unsigned)min(ncol[t], NB - 1);
  }

  const int k0 = blockIdx.x * KC;
  const int kend = min(k0 + KC, N);
  const int kal = k0 + ((kend - k0) & ~3);

  v8f acc[NTPW] = {};
  if (kal > k0) {                                      // pipelined main loop
    v2f a, b[NTPW];
    load_stage(S, B, abase, (unsigned)M, bbase, (unsigned)NB, k0, a, b);
#pragma unroll 2
    for (int kk = k0 + 4; kk < kal; kk += 4) {
      v2f an, bn[NTPW];
      load_stage(S, B, abase, (unsigned)M, bbase, (unsigned)NB, kk, an, bn);
      mma_stage(a, b, acc);
      a = an;
#pragma unroll
      for (int t = 0; t < NTPW; ++t) b[t] = bn[t];
    }
    mma_stage(a, b, acc);
  }
  if (kal < kend) {                                    // rare K tail (not hit here)
    const int kk = kal;
    v2f a;
#pragma unroll
    for (int v = 0; v < 2; ++v) {
      const int ka = kk + half * 2 + v;
      a[v] = (ka < kend) ? S[(unsigned)ka * (unsigned)M + (unsigned)min(mcol, M - 1)] : 0.0f;
    }
#pragma unroll
    for (int t = 0; t < NTPW; ++t) {
      v2f b;
#pragma unroll
      for (int v = 0; v < 2; ++v) {
        const int kb = kk + v + half * 2;
        b[v] = (kb < kend) ? B[(unsigned)kb * (unsigned)NB + (unsigned)min(ncol[t], NB - 1)] : 0.0f;
      }
      acc[t] = __builtin_amdgcn_wmma_f32_16x16x4_f32(false, a, false, b, (short)0,
                                                     acc[t], false, false);
    }
  }

#pragma unroll
  for (int t = 0; t < NTPW; ++t) {
    if (ncol[t] < NB) {
#pragma unroll
      for (int v = 0; v < 8; ++v) {
        const int m2 = mt * 16 + v + half * 8;
        if (m2 < M) atomicAdd(&OUT[(unsigned)m2 * (unsigned)NB + (unsigned)ncol[t]], acc[t][v]);
      }
    }
  }
}

// ---------------------------------------------------------------- launcher
static inline unsigned grid1d(long long work, long long cap = 32768) {
  long long b = (work + WG - 1) / WG;
  if (b < 1) b = 1;
  if (b > cap) b = cap;
  return (unsigned)b;
}

static void launch_scatter(const int* gidx, const int* sidx, const float* dinv,
                           const float* SRC, float* OUT, int E, int dim, int use_norm,
                           hipStream_t stream) {
  const unsigned blocks = grid1d((long long)E * (dim / 4));
  if (dim == 64)
    k_edge_scatter_t<16><<<blocks, WG, 0, stream>>>(gidx, sidx, dinv, SRC, OUT, E, use_norm);
  else if (dim == 500)
    k_edge_scatter_t<125><<<blocks, WG, 0, stream>>>(gidx, sidx, dinv, SRC, OUT, E, use_norm);
  else
    k_edge_scatter_g<<<blocks, WG, 0, stream>>>(gidx, sidx, dinv, SRC, OUT, E, dim, use_norm);
}

extern "C" void kernel_launch(void* const* d_in, const int* in_sizes, int n_in,
                              void* d_out, int out_size, void* d_ws, size_t ws_size,
                              hipStream_t stream) {
  const float* x        = (const float*)d_in[0];
  const int*   ei       = (const int*)d_in[1];     // [2,E]: row = ei[0..E), col = ei[E..2E)
  const float* W_embed  = (const float*)d_in[2];
  const float* b_embed  = (const float*)d_in[3];
  const float* W_assign = (const float*)d_in[4];
  const float* b_assign = (const float*)d_in[5];

  const int C = in_sizes[3];             // 64
  const int D = in_sizes[2] / C;         // 64
  const int N = in_sizes[0] / D;         // 50000
  const int E = in_sizes[1] / 2;         // 800000
  const int K = in_sizes[5];             // 500

  const int* erow = ei;
  const int* ecol = ei + E;

  // workspace carve (fp32): deg | dinv | H_embed | Z | Hbig(H_assign->AS) | Sbig(S)
  float* ws = (float*)d_ws;
  float* deg  = ws;  ws += N;
  float* dinv = ws;  ws += N;
  float* Hc   = ws;  ws += (size_t)N * C;
  float* Z    = ws;  ws += (size_t)N * C;
  float* Hk   = ws;  ws += (size_t)N * K;   // H_assign, later reused as AS
  float* Sm   = ws;  ws += (size_t)N * K;   // gcn_assign output -> softmax S

  float* nextX = (float*)d_out;                       // [K, C]
  float* nextA = (float*)d_out + (size_t)K * C;       // [K, K]

  // 1) degrees (self loop = init 1.0) and dinv
  k_fill<<<grid1d(N), WG, 0, stream>>>(deg, N, 1.0f);
  k_deg_edges<<<grid1d(E), WG, 0, stream>>>(erow, deg, E);
  k_dinv<<<grid1d(N), WG, 0, stream>>>(deg, dinv, N);

  // 2) linear transforms (fp32 WMMA)
  {
    const int mtiles = (N + 15) / 16;
    const int ngC = ((C + 15) / 16 + NTPW - 1) / NTPW;
    const int ngK = ((K + 15) / 16 + NTPW - 1) / NTPW;
    const long long unitsC = (long long)mtiles * ngC;
    const long long unitsK = (long long)mtiles * ngK;
    k_linear_wmma<<<(unsigned)((unitsC + 7) / 8), WG, 0, stream>>>(x, W_embed, b_embed, Hc, N, D, C, ngC);
    k_linear_wmma<<<(unsigned)((unitsK + 7) / 8), WG, 0, stream>>>(x, W_assign, b_assign, Hk, N, D, K, ngK);
  }

  // 3) GCN aggregation: self-loop term then edge scatter (gather row, scatter col)
  {
    const unsigned nb = grid1d((long long)N * 8);      // ~8 waves-worth of rows per block
    k_gcn_init<<<nb, WG, 0, stream>>>(Hc, dinv, Z, N, C);
    k_gcn_init<<<nb, WG, 0, stream>>>(Hk, dinv, Sm, N, K);
  }
  launch_scatter(erow, ecol, dinv, Hc, Z, E, C, 1, stream);
  launch_scatter(erow, ecol, dinv, Hk, Sm, E, K, 1, stream);

  // 4) softmax rows of assign output -> S
  k_softmax<<<(unsigned)((N + 7) / 8), WG, 0, stream>>>(Sm, N, K);

  // 5) AS = A @ S : AS[row] += S[col]; reuse Hk buffer
  k_fill<<<grid1d((long long)N * K), WG, 0, stream>>>(Hk, (long long)N * K, 0.0f);
  launch_scatter(ecol, erow, dinv, Sm, Hk, E, K, 0, stream);

  // 6) outputs: zero then split-K WMMA reductions
  k_fill<<<grid1d(out_size), WG, 0, stream>>>((float*)d_out, out_size, 0.0f);
  {
    const int KC = 2048;
    const int chunks = (N + KC - 1) / KC;
    const int mtiles = (K + 15) / 16;
    const int ngC = ((C + 15) / 16 + NTPW - 1) / NTPW;
    const int ngK = ((K + 15) / 16 + NTPW - 1) / NTPW;
    dim3 gX((unsigned)chunks, (unsigned)(((long long)mtiles * ngC + 7) / 8));
    dim3 gA((unsigned)chunks, (unsigned)(((long long)mtiles * ngK + 7) / 8));
    k_spt_gemm_wmma<<<gX, WG, 0, stream>>>(Sm, Z,  nextX, N, K, C, KC, ngC);  // next_X = S^T Z
    k_spt_gemm_wmma<<<gA, WG, 0, stream>>>(Sm, Hk, nextA, N, K, K, KC, ngK);  // next_A = S^T (A S)
  }
}